// ChebConv_46153718563236
// MI455X (gfx1250) — compile-verified
//
#include <hip/hip_runtime.h>

#define NN 100000
#define NE 1250000
#define D  64
// K=3, LAMBDA_MAX=2.0 -> re_norm = 1.0 exactly, so:
//   X1 = -L(X0)
//   X2 = -2*L(X1) - X0

typedef __attribute__((ext_vector_type(2))) float v2f;
typedef __attribute__((ext_vector_type(8))) float v8f;

__global__ void zero_f32(float* __restrict__ p, int n) {
  int i = blockIdx.x * blockDim.x + threadIdx.x;
  if (i < n) p[i] = 0.0f;
}

__global__ void deg_count(const int* __restrict__ dst, float* __restrict__ deg, int E) {
  int e = blockIdx.x * blockDim.x + threadIdx.x;
  if (e < E) atomicAdd(&deg[dst[e]], 1.0f);
}

__global__ void make_dsqrt(float* __restrict__ deg, int n) {
  int i = blockIdx.x * blockDim.x + threadIdx.x;
  if (i < n) {
    float d = fmaxf(deg[i], 1.0f);
    deg[i] = 1.0f / sqrtf(d);
  }
}

// agg[dst] += x[src] * dsqrt[src] ; one thread per (edge, 4-channel group)
__global__ __launch_bounds__(256)
void edge_scatter(const float* __restrict__ x, const float* __restrict__ dsqrt,
                  const int* __restrict__ src, const int* __restrict__ dst,
                  float* __restrict__ agg, int E) {
  int tid = blockIdx.x * blockDim.x + threadIdx.x;
  int e = tid >> 4;
  int g = (tid & 15) << 2;
  if (e >= E) return;
  int s = src[e];
  int d = dst[e];
  float ds = dsqrt[s];
  float4 v = *reinterpret_cast<const float4*>(x + (size_t)s * D + g);
  float* a = agg + (size_t)d * D + g;
  atomicAdd(a + 0, v.x * ds);
  atomicAdd(a + 1, v.y * ds);
  atomicAdd(a + 2, v.z * ds);
  atomicAdd(a + 3, v.w * ds);
}

// X1 = -(agg * dsqrt[row]); also re-zero agg for the next propagation
__global__ __launch_bounds__(256)
void combine_x1(const float* __restrict__ dsqrt, float* __restrict__ agg,
                float* __restrict__ x1, int n4) {
  int i = blockIdx.x * blockDim.x + threadIdx.x;
  if (i >= n4) return;
  int row = i >> 4;  // 16 float4 per row of 64
  float ds = dsqrt[row];
  float4* ap = reinterpret_cast<float4*>(agg) + i;
  float4 a = *ap;
  float4 r;
  r.x = -a.x * ds; r.y = -a.y * ds; r.z = -a.z * ds; r.w = -a.w * ds;
  reinterpret_cast<float4*>(x1)[i] = r;
  float4 z; z.x = 0.f; z.y = 0.f; z.z = 0.f; z.w = 0.f;
  *ap = z;
}

// X2 = -2*(agg * dsqrt[row]) - X0
__global__ __launch_bounds__(256)
void combine_x2(const float* __restrict__ dsqrt, const float* __restrict__ agg,
                const float* __restrict__ x0, float* __restrict__ x2, int n4) {
  int i = blockIdx.x * blockDim.x + threadIdx.x;
  if (i >= n4) return;
  int row = i >> 4;
  float ds2 = -2.0f * dsqrt[row];
  float4 a = reinterpret_cast<const float4*>(agg)[i];
  float4 f = reinterpret_cast<const float4*>(x0)[i];
  float4 r;
  r.x = fmaf(a.x, ds2, -f.x);
  r.y = fmaf(a.y, ds2, -f.y);
  r.z = fmaf(a.z, ds2, -f.z);
  r.w = fmaf(a.w, ds2, -f.w);
  reinterpret_cast<float4*>(x2)[i] = r;
}

// out[N,64] = relu(X0@W[0:64] + X1@W[64:128] + X2@W[128:192] + b)
// 8 waves per block; each wave: 16 rows x 64 cols via 4 accumulator tiles,
// 192 v_wmma_f32_16x16x4_f32 per wave. W staged in LDS (48 KB).
__global__ __launch_bounds__(256)
void cheb_gemm(const float* __restrict__ x0, const float* __restrict__ x1,
               const float* __restrict__ x2, const float* __restrict__ W,
               const float* __restrict__ bias, float* __restrict__ out, int n) {
  __shared__ float Wl[3 * D * D];  // [192][64], 48 KB
  {
    const float4* Wg = reinterpret_cast<const float4*>(W);
    float4* Ws = reinterpret_cast<float4*>(Wl);
    for (int i = threadIdx.x; i < 3 * D * D / 4; i += 256) Ws[i] = Wg[i];
  }
  __syncthreads();

  const int wave = threadIdx.x >> 5;
  const int lane = threadIdx.x & 31;
  const int half = lane >> 4;   // 0: K pair {0,1}, 1: K pair {2,3}
  const int m    = lane & 15;   // A: row-in-tile, B: col-in-tile, D: col-in-tile
  const int row0 = (blockIdx.x * 8 + wave) * 16;
  if (row0 >= n) return;  // wave-uniform: EXEC stays all-ones for WMMA

  v8f c0 = {}, c1 = {}, c2 = {}, c3 = {};
  const float* Xs[3] = {x0, x1, x2};

  for (int t = 0; t < 3; ++t) {
    const float* arow = Xs[t] + (size_t)(row0 + m) * D;
    const float* wblk = Wl + t * D * D;
#pragma unroll
    for (int kk = 0; kk < D; kk += 4) {
      const int ka = kk + 2 * half;  // this half-wave's K pair
      v2f a;
      a.x = arow[ka];
      a.y = arow[ka + 1];
      const float* w0 = wblk + ka * D;  // W row ka
      const float* w1 = w0 + D;         // W row ka+1
      v2f b0, b1, b2, b3;
      b0.x = w0[m];      b0.y = w1[m];
      b1.x = w0[16 + m]; b1.y = w1[16 + m];
      b2.x = w0[32 + m]; b2.y = w1[32 + m];
      b3.x = w0[48 + m]; b3.y = w1[48 + m];
      c0 = __builtin_amdgcn_wmma_f32_16x16x4_f32(false, a, false, b0, (short)0, c0, false, false);
      c1 = __builtin_amdgcn_wmma_f32_16x16x4_f32(false, a, false, b1, (short)0, c1, false, false);
      c2 = __builtin_amdgcn_wmma_f32_16x16x4_f32(false, a, false, b2, (short)0, c2, false, false);
      c3 = __builtin_amdgcn_wmma_f32_16x16x4_f32(false, a, false, b3, (short)0, c3, false, false);
    }
  }

  const float bb0 = bias[m], bb1 = bias[16 + m], bb2 = bias[32 + m], bb3 = bias[48 + m];
#pragma unroll
  for (int v = 0; v < 8; ++v) {
    // C/D layout: VGPR v, lanes 0-15 -> row v, col=lane; lanes 16-31 -> row v+8
    int row = row0 + v + 8 * half;
    float* o = out + (size_t)row * D;
    o[m]      = fmaxf(c0[v] + bb0, 0.0f);
    o[16 + m] = fmaxf(c1[v] + bb1, 0.0f);
    o[32 + m] = fmaxf(c2[v] + bb2, 0.0f);
    o[48 + m] = fmaxf(c3[v] + bb3, 0.0f);
  }
}

extern "C" void kernel_launch(void* const* d_in, const int* in_sizes, int n_in,
                              void* d_out, int out_size, void* d_ws, size_t ws_size,
                              hipStream_t stream) {
  const float* feat = (const float*)d_in[0];
  const int*   src  = (const int*)d_in[1];
  const int*   dst  = (const int*)d_in[2];
  const float* W    = (const float*)d_in[3];
  const float* bias = (const float*)d_in[4];
  float* out = (float*)d_out;

  // workspace layout (all 16B-aligned): dsqrt[N], agg[N*D], x1[N*D], x2[N*D]
  float* dsqrt = (float*)d_ws;
  float* agg   = dsqrt + NN;
  float* x1    = agg + (size_t)NN * D;
  float* x2    = x1 + (size_t)NN * D;

  const int B = 256;
  zero_f32<<<(NN + B - 1) / B, B, 0, stream>>>(dsqrt, NN);
  zero_f32<<<(NN * D + B - 1) / B, B, 0, stream>>>(agg, NN * D);
  deg_count<<<(NE + B - 1) / B, B, 0, stream>>>(dst, dsqrt, NE);
  make_dsqrt<<<(NN + B - 1) / B, B, 0, stream>>>(dsqrt, NN);

  const int scatterBlocks = (NE * 16 + B - 1) / B;  // 78125
  edge_scatter<<<scatterBlocks, B, 0, stream>>>(feat, dsqrt, src, dst, agg, NE);
  combine_x1<<<(NN * 16 + B - 1) / B, B, 0, stream>>>(dsqrt, agg, x1, NN * 16);
  edge_scatter<<<scatterBlocks, B, 0, stream>>>(x1, dsqrt, src, dst, agg, NE);
  combine_x2<<<(NN * 16 + B - 1) / B, B, 0, stream>>>(dsqrt, agg, feat, x2, NN * 16);

  cheb_gemm<<<(NN + 127) / 128, 256, 0, stream>>>(feat, x1, x2, W, bias, out, NN);
}